// PrototypicalLoss_38362647888584
// MI455X (gfx1250) — compile-verified
//
#include <hip/hip_runtime.h>
#include <math.h>

// Problem constants (from setup_inputs: n_way=64, n_support=16, n_query=64, d=1024)
#define N_WAY   64
#define N_SUP   16
#define N_QRY   64
#define GROUP   (N_SUP + N_QRY)   // 80 rows per class in x
#define DIM     1024
#define N_QUERIES (N_WAY * N_QRY) // 4096

typedef __attribute__((ext_vector_type(2))) float v2f;
typedef __attribute__((ext_vector_type(8))) float v8f;

// ---------------------------------------------------------------------------
// Kernel 1: prototypes P[w, d] = mean over 16 support rows; pp[w] = ||P[w]||^2
// grid = 64 (one class per block), block = 256 (4 consecutive floats/thread)
// ---------------------------------------------------------------------------
__global__ void proto_kernel(const float* __restrict__ x,
                             float* __restrict__ P,
                             float* __restrict__ pp,
                             float* __restrict__ accum) {
    const int w   = blockIdx.x;
    const int tid = threadIdx.x;

    // zero the loss/correct accumulators (stream-ordered before finalize)
    if (w == 0 && tid < 2) accum[tid] = 0.0f;

    const int d = tid * 4;
    const float* base = x + (size_t)w * GROUP * DIM + d;

    float4 s = make_float4(0.f, 0.f, 0.f, 0.f);
#pragma unroll
    for (int srow = 0; srow < N_SUP; ++srow) {
        const float4 v = *(const float4*)(base + (size_t)srow * DIM);
        s.x += v.x; s.y += v.y; s.z += v.z; s.w += v.w;
    }
    const float inv = 1.0f / (float)N_SUP;
    s.x *= inv; s.y *= inv; s.z *= inv; s.w *= inv;
    *(float4*)(P + (size_t)w * DIM + d) = s;

    float sq = s.x * s.x + s.y * s.y + s.z * s.z + s.w * s.w;

    __shared__ float red[256];
    red[tid] = sq;
    __syncthreads();
#pragma unroll
    for (int off = 128; off > 0; off >>= 1) {
        if (tid < off) red[tid] += red[tid + off];
        __syncthreads();
    }
    if (tid == 0) pp[w] = red[0];
}

// ---------------------------------------------------------------------------
// Kernel 2: WMMA GEMM.  logits[n, w] = 2 * dot(q_n, p_w) - ||p_w||^2
// (||q_n||^2 dropped: constant per row -> invariant under log_softmax/argmax)
// One wave computes one 16x16 tile via V_WMMA_F32_16X16X4_F32, K-loop 256x.
// grid = 128 blocks x 256 threads = 1024 waves = 256 Mtiles x 4 Ntiles.
// ---------------------------------------------------------------------------
__global__ void gemm_kernel(const float* __restrict__ x,
                            const float* __restrict__ P,
                            const float* __restrict__ pp,
                            float* __restrict__ logits) {
    const int wave  = (blockIdx.x * blockDim.x + threadIdx.x) >> 5; // 0..1023
    const int lane  = threadIdx.x & 31;
    const int tileM = wave >> 2;   // 0..255 (queries, 16 each)
    const int tileN = wave & 3;    // 0..3   (classes, 16 each)

    // A (16x4 f32) layout: lane L<16 holds row M=L, K = {0,1}; lane L>=16 same
    // row, K = {2,3}.  B (4x16) mirrors: lane%16 selects class row, same koff.
    const int row  = lane & 15;
    const int koff = (lane >> 4) << 1;   // 0 or 2

    const int n      = tileM * 16 + row;                    // query index
    const int q_row  = (n >> 6) * GROUP + N_SUP + (n & 63); // row in x
    const float* qptr = x + (size_t)q_row * DIM + koff;
    const float* bptr = P + (size_t)(tileN * 16 + row) * DIM + koff;

    v8f c = {};
    for (int k = 0; k < DIM; k += 4) {
        v2f a = *(const v2f*)(qptr + k);
        v2f b = *(const v2f*)(bptr + k);
        // D = A(16x4, f32) x B(4x16, f32) + C(16x16, f32)
        c = __builtin_amdgcn_wmma_f32_16x16x4_f32(
            /*neg_a=*/false, a, /*neg_b=*/false, b,
            /*c_mod=*/(short)0, c, /*reuse_a=*/false, /*reuse_b=*/false);
    }

    // C/D layout: lane<16 -> (M=r, N=lane); lane>=16 -> (M=r+8, N=lane-16)
    const int nCol  = tileN * 16 + (lane & 15);
    const int mBase = tileM * 16 + ((lane >> 4) << 3);
    const float ppv = pp[nCol];
#pragma unroll
    for (int r = 0; r < 8; ++r) {
        logits[(size_t)(mBase + r) * N_WAY + nCol] = 2.0f * c[r] - ppv;
    }
}

// ---------------------------------------------------------------------------
// Kernel 3: per-row log-softmax at the true label + argmax, wave per row.
// 64 logits/row -> 2 per lane (wave32).  grid = 512 x 256 (8 waves/block).
// ---------------------------------------------------------------------------
__global__ void finalize_kernel(const float* __restrict__ logits,
                                float* __restrict__ accum) {
    const int lane = threadIdx.x & 31;
    const int rown = blockIdx.x * 8 + (threadIdx.x >> 5); // 0..4095
    const int label = rown >> 6;                          // n = w*64 + j

    const float v0 = logits[(size_t)rown * N_WAY + lane];
    const float v1 = logits[(size_t)rown * N_WAY + lane + 32];

    // --- combined max + argmax (first index wins ties, matching jnp.argmax)
    float bv = v0; int bi = lane;
    if (v1 > bv) { bv = v1; bi = lane + 32; }
#pragma unroll
    for (int off = 16; off > 0; off >>= 1) {
        float ov = __shfl_xor(bv, off, 32);
        int   oi = __shfl_xor(bi, off, 32);
        if (ov > bv || (ov == bv && oi < bi)) { bv = ov; bi = oi; }
    }
    const float rmax = bv;

    // --- stable logsumexp
    float ssum = __expf(v0 - rmax) + __expf(v1 - rmax);
#pragma unroll
    for (int off = 16; off > 0; off >>= 1)
        ssum += __shfl_xor(ssum, off, 32);
    const float lse = rmax + __logf(ssum);

    // --- logit at the true label (broadcast from the owning lane)
    const float ll = (label < 32) ? __shfl(v0, label, 32)
                                  : __shfl(v1, label - 32, 32);

    if (lane == 0) {
        atomicAdd(&accum[0], lse - ll);                      // -logp[label]
        atomicAdd(&accum[1], (bi == label) ? 1.0f : 0.0f);   // correct count
    }
}

__global__ void writeout_kernel(const float* __restrict__ accum,
                                float* __restrict__ out) {
    out[0] = accum[0] * (1.0f / (float)N_QUERIES);
    out[1] = accum[1] * (100.0f / (float)N_QUERIES);
}

// ---------------------------------------------------------------------------
extern "C" void kernel_launch(void* const* d_in, const int* in_sizes, int n_in,
                              void* d_out, int out_size, void* d_ws, size_t ws_size,
                              hipStream_t stream) {
    (void)in_sizes; (void)n_in; (void)out_size; (void)ws_size;
    const float* x = (const float*)d_in[0];   // [5120, 1024] f32
    float* out = (float*)d_out;               // [loss, acc]

    // Workspace layout (floats): P[64*1024] | pp[64] | logits[4096*64] | accum[2]
    float* P      = (float*)d_ws;
    float* pp     = P + (size_t)N_WAY * DIM;
    float* logits = pp + N_WAY;
    float* accum  = logits + (size_t)N_QUERIES * N_WAY;

    proto_kernel   <<<N_WAY, 256, 0, stream>>>(x, P, pp, accum);
    gemm_kernel    <<<128,   256, 0, stream>>>(x, P, pp, logits);
    finalize_kernel<<<N_QUERIES / 8, 256, 0, stream>>>(logits, accum);
    writeout_kernel<<<1, 1, 0, stream>>>(accum, out);
}